// Quantize_10084583211046
// MI455X (gfx1250) — compile-verified
//
#include <hip/hip_runtime.h>

typedef __attribute__((ext_vector_type(2))) float v2f;
typedef __attribute__((ext_vector_type(8))) float v8f;

#define DIM    512
#define NEMB   4096
#define NROWS  (16 * 2048)      // 32768 flattened tokens
#define MTILE  64               // rows of x staged in LDS per block (128 KB)
#define NWAVES 4                // waves per block

// ---------------------------------------------------------------------------
// Kernel 1: invnorm[j] = 1 / ||embed_weight[j, :]||   (4096 rows of 512)
// ---------------------------------------------------------------------------
__global__ __launch_bounds__(128) void rownorm_kernel(const float* __restrict__ emb,
                                                      float* __restrict__ invnorm) {
    __shared__ float red[128];
    const int row = blockIdx.x;
    const float* r = emb + (size_t)row * DIM;
    float s = 0.0f;
    for (int i = threadIdx.x; i < DIM; i += 128) {
        float v = r[i];
        s += v * v;
    }
    red[threadIdx.x] = s;
    __syncthreads();
    for (int off = 64; off > 0; off >>= 1) {
        if (threadIdx.x < off) red[threadIdx.x] += red[threadIdx.x + off];
        __syncthreads();
    }
    if (threadIdx.x == 0) invnorm[row] = rsqrtf(red[0]);
}

// ---------------------------------------------------------------------------
// Kernel 2: fused GEMM + row-argmax via V_WMMA_F32_16X16X4_F32.
//   block = 128 threads (4 waves), A tile = 64 rows x 512 staged in LDS
//   (128 KB of the WGP's 320 KB). 4-way M-register-blocking: each wave
//   multiplies all four 16-row A sub-tiles against one B tile (4 v8f
//   accumulators), so every B float2 load feeds FOUR WMMAs -> L2 B-traffic
//   drops to ~4 GB total. The 4 waves split the 256 N-tiles; per-row argmax
//   candidates are merged through LDS with first-index tie-breaking.
// ---------------------------------------------------------------------------
__global__ __launch_bounds__(128) void argmax_gemm_kernel(const float* __restrict__ x,
                                                          const float* __restrict__ emb,
                                                          const float* __restrict__ invnorm,
                                                          int* __restrict__ ind) {
    __shared__ float lA[MTILE * DIM];            // 64*512*4 = 128 KB

    const int tid  = threadIdx.x;
    const int lane = tid & 31;
    const int wave = tid >> 5;                   // 0..3
    const int rowBlock = blockIdx.x * MTILE;

    // cooperative A-tile load: 64*512 floats, float4 strided across 128 threads
    {
        const float4* src = (const float4*)(x + (size_t)rowBlock * DIM);
        float4* dst = (float4*)lA;
        for (int i = tid; i < MTILE * DIM / 4; i += 128) dst[i] = src[i];
    }
    __syncthreads();

    const int lmod  = lane & 15;
    const int lhalf = lane >> 4;                 // 0 or 1
    const int khalf = lhalf * 2;                 // K offset within a K4 step

    const float* aRow[4];
#pragma unroll
    for (int q = 0; q < 4; ++q)
        aRow[q] = lA + (size_t)(q * 16 + lmod) * DIM;

    float best[4][8];
    int   bidx[4][8];
#pragma unroll
    for (int q = 0; q < 4; ++q)
#pragma unroll
        for (int v = 0; v < 8; ++v) { best[q][v] = -3.402823466e38f; bidx[q][v] = 0; }

    // waves split the N-tile range
    for (int nt = wave; nt < NEMB / 16; nt += NWAVES) {
        const int col = nt * 16 + lmod;          // this lane's output column
        const float* bRow = emb + (size_t)col * DIM;

        v8f c[4];
#pragma unroll
        for (int q = 0; q < 4; ++q)
            c[q] = (v8f){0.f, 0.f, 0.f, 0.f, 0.f, 0.f, 0.f, 0.f};

#pragma unroll 2
        for (int k0 = 0; k0 < DIM; k0 += 4) {
            const int kk = k0 + khalf;
            // B lane layout: lane L holds N=L%16, K pair {khalf, khalf+1}
            const v2f b = *(const v2f*)(bRow + kk);
#pragma unroll
            for (int q = 0; q < 4; ++q) {
                // A lane layout: lane L holds M=L%16, same K pair
                const v2f a = *(const v2f*)(aRow[q] + kk);
                c[q] = __builtin_amdgcn_wmma_f32_16x16x4_f32(
                           false, a, false, b, (short)0, c[q], false, false);
            }
        }

        const float s = invnorm[col];
#pragma unroll
        for (int q = 0; q < 4; ++q)
#pragma unroll
            for (int v = 0; v < 8; ++v) {
                const float sc = c[q][v] * s;
                if (sc > best[q][v]) { best[q][v] = sc; bidx[q][v] = col; } // strict > keeps lowest col
            }
    }

    // merge the 16 column-residue lanes of each row group (first-index ties)
#pragma unroll
    for (int m = 1; m < 16; m <<= 1) {
#pragma unroll
        for (int q = 0; q < 4; ++q)
#pragma unroll
            for (int v = 0; v < 8; ++v) {
                const float ob = __shfl_xor(best[q][v], m, 32);
                const int   oi = __shfl_xor(bidx[q][v], m, 32);
                if (ob > best[q][v] || (ob == best[q][v] && oi < bidx[q][v])) {
                    best[q][v] = ob; bidx[q][v] = oi;
                }
            }
    }

    // cross-wave merge through LDS (reuse A-tile storage after a barrier).
    __syncthreads();                              // all waves done reading lA
    float* bestBuf = lA;                          // [NWAVES][MTILE] floats
    int*   idxBuf  = ((int*)lA) + NWAVES * MTILE; // [NWAVES][MTILE] ints

    // C/D layout: VGPR v holds rows M = q*16 + lhalf*8 + v for lanes lmod==0
    if (lmod == 0) {
#pragma unroll
        for (int q = 0; q < 4; ++q) {
            const int base = wave * MTILE + q * 16 + lhalf * 8;
#pragma unroll
            for (int v = 0; v < 8; ++v) {
                bestBuf[base + v] = best[q][v];
                idxBuf[base + v]  = bidx[q][v];
            }
        }
    }
    __syncthreads();

    if (tid < MTILE) {
        float bs = bestBuf[tid];
        int   bi = idxBuf[tid];
#pragma unroll
        for (int w = 1; w < NWAVES; ++w) {
            const float s = bestBuf[w * MTILE + tid];
            const int   i = idxBuf[w * MTILE + tid];
            if (s > bs || (s == bs && i < bi)) { bs = s; bi = i; }
        }
        ind[rowBlock + tid] = bi;
    }
}

// ---------------------------------------------------------------------------
// Kernel 3: gather quantize rows + per-row squared-error partial sums.
// ---------------------------------------------------------------------------
__global__ __launch_bounds__(128) void gather_diff_kernel(const float* __restrict__ x,
                                                          const float* __restrict__ emb,
                                                          const int* __restrict__ ind,
                                                          float* __restrict__ out,
                                                          float* __restrict__ partials) {
    __shared__ float red[128];
    const int n = blockIdx.x;
    const int idx = ind[n];
    const float* e  = emb + (size_t)idx * DIM;
    const float* xr = x   + (size_t)n   * DIM;
    float*       o  = out + (size_t)n   * DIM;

    float s = 0.0f;
    for (int i = threadIdx.x; i < DIM; i += 128) {
        const float q = e[i];
        const float d = q - xr[i];
        o[i] = q;                                 // (q + q) * 0.5 == q
        s += d * d;
    }
    red[threadIdx.x] = s;
    __syncthreads();
    for (int off = 64; off > 0; off >>= 1) {
        if (threadIdx.x < off) red[threadIdx.x] += red[threadIdx.x + off];
        __syncthreads();
    }
    if (threadIdx.x == 0) partials[n] = red[0];
}

// ---------------------------------------------------------------------------
// Kernel 4: deterministic final reduction of 32768 partials -> diff scalar.
// ---------------------------------------------------------------------------
__global__ __launch_bounds__(1024) void finalize_kernel(const float* __restrict__ partials,
                                                        float* __restrict__ diff_out) {
    __shared__ float red[1024];
    float s = 0.0f;
    for (int i = threadIdx.x; i < NROWS; i += 1024) s += partials[i];
    red[threadIdx.x] = s;
    __syncthreads();
    for (int off = 512; off > 0; off >>= 1) {
        if (threadIdx.x < off) red[threadIdx.x] += red[threadIdx.x + off];
        __syncthreads();
    }
    if (threadIdx.x == 0)
        diff_out[0] = red[0] / (float)((size_t)NROWS * DIM);
}

// ---------------------------------------------------------------------------
extern "C" void kernel_launch(void* const* d_in, const int* in_sizes, int n_in,
                              void* d_out, int out_size, void* d_ws, size_t ws_size,
                              hipStream_t stream) {
    const float* x   = (const float*)d_in[0];   // [16, 2048, 512] f32
    const float* emb = (const float*)d_in[1];   // [4096, 512]      f32
    float* out = (float*)d_out;                 // [16*2048*512] quantize, then diff

    // workspace layout (floats): invnorm[4096] | ind[32768] (as int) | partials[32768]
    float* invnorm  = (float*)d_ws;
    int*   ind      = (int*)((char*)d_ws + (size_t)NEMB * 4);
    float* partials = (float*)((char*)d_ws + (size_t)(NEMB + NROWS) * 4);

    rownorm_kernel<<<NEMB, 128, 0, stream>>>(emb, invnorm);
    argmax_gemm_kernel<<<NROWS / MTILE, 128, 0, stream>>>(x, emb, invnorm, ind);
    gather_diff_kernel<<<NROWS, 128, 0, stream>>>(x, emb, ind, out, partials);
    finalize_kernel<<<1, 1024, 0, stream>>>(partials, out + (size_t)NROWS * DIM);
}